// SinkhornKnopp_28965259444505
// MI455X (gfx1250) — compile-verified
//
#include <hip/hip_runtime.h>
#include <hip/hip_bf16.h>

#define NN 8192
#define SK_EPS 1e-8f
#define ITERS 20

typedef __attribute__((ext_vector_type(2))) float v2f;
typedef __attribute__((ext_vector_type(8))) float v8f;

// ---------------------------------------------------------------------------
// CDNA5 async global->LDS helpers (ASYNCcnt path).  The builtin's pointer
// params are addrspace-qualified pointers to 16-byte int vectors (per hipcc
// diagnostic), so cast through those exactly.
// ---------------------------------------------------------------------------
#if __has_builtin(__builtin_amdgcn_global_load_async_to_lds_b128)
#define HAS_ASYNC_LDS 1
#else
#define HAS_ASYNC_LDS 0
#endif

typedef int vb128_t __attribute__((vector_size(16)));
typedef __attribute__((address_space(1))) vb128_t* gas_b128_ptr;
typedef __attribute__((address_space(3))) vb128_t* las_b128_ptr;

__device__ __forceinline__ void async_copy16(const float* gsrc, float* ldst) {
#if HAS_ASYNC_LDS
    __builtin_amdgcn_global_load_async_to_lds_b128(
        (gas_b128_ptr)gsrc, (las_b128_ptr)ldst, 0, 0);
#else
    *(float4*)ldst = *(const float4*)gsrc;
#endif
}

__device__ __forceinline__ void async_wait0() {
#if HAS_ASYNC_LDS
#if __has_builtin(__builtin_amdgcn_s_wait_asynccnt)
    __builtin_amdgcn_s_wait_asynccnt(0);
#else
    asm volatile("s_wait_asynccnt 0" ::: "memory");
#endif
#endif
}

// ---------------------------------------------------------------------------
// Pass 1: P0 = exp(M - rowmax(M)).  One block per row; row staged in LDS so the
// matrix is read only once for both the max-reduction and the exp.
// ---------------------------------------------------------------------------
__global__ __launch_bounds__(256) void rowexp_kernel(const float* __restrict__ M,
                                                     float* __restrict__ P) {
    __shared__ float row[NN];     // 32 KB
    __shared__ float red[256];
    const int tid = threadIdx.x;
    const size_t base = (size_t)blockIdx.x * NN;

    float lmax = -3.402823466e38f;
#pragma unroll
    for (int q = 0; q < 8; ++q) {
        const int col = q * 1024 + tid * 4;
        const float4 v = *(const float4*)(M + base + col);
        *(float4*)(row + col) = v;
        lmax = fmaxf(lmax, fmaxf(fmaxf(v.x, v.y), fmaxf(v.z, v.w)));
    }
    red[tid] = lmax;
    __syncthreads();
    for (int o = 128; o > 0; o >>= 1) {
        if (tid < o) red[tid] = fmaxf(red[tid], red[tid + o]);
        __syncthreads();
    }
    const float rmax = red[0];
#pragma unroll
    for (int q = 0; q < 8; ++q) {
        const int col = q * 1024 + tid * 4;
        const float4 v = *(const float4*)(row + col);
        float4 e;
        e.x = __expf(v.x - rmax);
        e.y = __expf(v.y - rmax);
        e.z = __expf(v.z - rmax);
        e.w = __expf(v.w - rmax);
        *(float4*)(P + base + col) = e;
    }
}

// ---------------------------------------------------------------------------
// y = P0 * c   via V_WMMA_F32_16X16X4_F32 with double-buffered async
// global->LDS staging: while WMMAs consume tile t, the async DMA fills tile
// t+1 (tracked by ASYNCcnt, drained with s_wait_asynccnt before the barrier).
// A fragment: lanes 0-15 hold M=0..15, lane-half selects K{0,1}/{2,3}.
// B broadcasts c[k] across all 16 columns so every column of D accumulates
// the c-weighted row sum.
// ---------------------------------------------------------------------------
__global__ __launch_bounds__(256) void rowmatvec_wmma(const float* __restrict__ P,
                                                      const float* __restrict__ c,
                                                      float* __restrict__ y) {
    constexpr int TILE = 256;
    constexpr int LST  = TILE + 4;        // 260 % 64 == 4 -> rows in distinct banks
    constexpr int NT   = NN / TILE;       // 32 tiles
    __shared__ float tile[2][16 * LST];   // ~33 KB double buffer
    __shared__ float wsum[8][16];

    const int tid  = threadIdx.x;
    const int wave = tid >> 5;
    const int lane = tid & 31;
    const int m    = lane & 15;           // M index of A fragment
    const int h    = lane >> 4;           // half: selects K pair
    const int rowbase = blockIdx.x * 16;

    const int lrow = tid >> 4;            // loader: row 0..15
    const int lcol = (tid & 15) * 4;      // loader: float4 base in 64-col chunk

    const float* srow = P + (size_t)(rowbase + lrow) * NN;

    v8f acc = {};

    // Prologue: stage tile 0.
#pragma unroll
    for (int q = 0; q < TILE / 64; ++q)
        async_copy16(srow + lcol + q * 64, &tile[0][lrow * LST] + lcol + q * 64);
    async_wait0();
    __syncthreads();

    for (int t = 0; t < NT; ++t) {
        const int buf = t & 1;
        // Kick off the DMA for the next tile into the other buffer.
        if (t + 1 < NT) {
            const float* src = srow + (t + 1) * TILE;
            float* dst = &tile[buf ^ 1][lrow * LST];
#pragma unroll
            for (int q = 0; q < TILE / 64; ++q)
                async_copy16(src + lcol + q * 64, dst + lcol + q * 64);
        }

        // WMMA on the current tile (overlaps with the async loads above).
        const int colbase = t * TILE;
        const int wbase = wave * (TILE / 8);   // 32-col strip per wave
#pragma unroll
        for (int s = 0; s < TILE / 32; ++s) {  // 8 K-steps of 4
            const int k0 = wbase + s * 4;
            v2f a, b;
            const float* ap = &tile[buf][m * LST + k0 + 2 * h];
            a.x = ap[0];
            a.y = ap[1];
            const float* cp = c + colbase + k0 + 2 * h;
            b.x = cp[0];
            b.y = cp[1];
            acc = __builtin_amdgcn_wmma_f32_16x16x4_f32(
                false, a, false, b, (short)0, acc, false, false);
        }

        async_wait0();      // next tile fully landed in LDS
        __syncthreads();
    }

    // D[i][j] identical across j; lanes with N==0 (lane 0, lane 16) hold rows
    // v + 8*h in acc[v].
    if ((lane & 15) == 0) {
#pragma unroll
        for (int v = 0; v < 8; ++v) wsum[wave][v + 8 * h] = acc[v];
    }
    __syncthreads();
    if (tid < 16) {
        float s = 0.f;
#pragma unroll
        for (int w = 0; w < 8; ++w) s += wsum[w][tid];
        y[rowbase + tid] = s;
    }
}

// ---------------------------------------------------------------------------
// z += P0^T * r, slab-parallel.  Block = 256 threads covers 1024 columns
// (float4 per thread, fully coalesced) x 256-row slab; r slab staged in LDS;
// slab partials combined with GLOBAL_ATOMIC_ADD_F32.  Prefetch runs 8 rows
// ahead of the FMA stream (global_prefetch_b8).
// ---------------------------------------------------------------------------
__global__ __launch_bounds__(256) void colmatvec(const float* __restrict__ P,
                                                 const float* __restrict__ r,
                                                 float* __restrict__ z) {
    __shared__ float rs[256];
    const int tid = threadIdx.x;
    const int j0 = blockIdx.x * 1024 + tid * 4;
    const int i0 = blockIdx.y * 256;
    rs[tid] = r[i0 + tid];
    __syncthreads();

    float4 acc = {0.f, 0.f, 0.f, 0.f};
    for (int ii = 0; ii < 256; ++ii) {
        const float* p = P + (size_t)(i0 + ii) * NN + j0;
        if (ii + 8 < 256)
            __builtin_prefetch(P + (size_t)(i0 + ii + 8) * NN + j0, 0, 1);
        const float4 v = *(const float4*)p;
        const float rv = rs[ii];
        acc.x = fmaf(v.x, rv, acc.x);
        acc.y = fmaf(v.y, rv, acc.y);
        acc.z = fmaf(v.z, rv, acc.z);
        acc.w = fmaf(v.w, rv, acc.w);
    }
    atomicAdd(&z[j0 + 0], acc.x);
    atomicAdd(&z[j0 + 1], acc.y);
    atomicAdd(&z[j0 + 2], acc.z);
    atomicAdd(&z[j0 + 3], acc.w);
}

// ---------------------------------------------------------------------------
// Small vector kernels.
// ---------------------------------------------------------------------------
__global__ void init_rc(float* __restrict__ r, float* __restrict__ c) {
    const int i = blockIdx.x * 256 + threadIdx.x;
    r[i] = 1.f;
    c[i] = 1.f;
}

__global__ void update_r_zero_z(float* __restrict__ r, const float* __restrict__ y,
                                float* __restrict__ z) {
    const int i = blockIdx.x * 256 + threadIdx.x;
    const float rv = r[i];
    r[i] = rv / (rv * y[i] + SK_EPS);   // r / (rowsum + eps)
    z[i] = 0.f;                          // clear colsum accumulator
}

__global__ void update_c(float* __restrict__ c, const float* __restrict__ z) {
    const int j = blockIdx.x * 256 + threadIdx.x;
    const float cv = c[j];
    c[j] = cv / (cv * z[j] + SK_EPS);   // c / (colsum + eps)
}

// ---------------------------------------------------------------------------
// Final: P = diag(r) * P0 * diag(c), in place over d_out.
// ---------------------------------------------------------------------------
__global__ __launch_bounds__(256) void final_scale(float* __restrict__ P,
                                                   const float* __restrict__ r,
                                                   const float* __restrict__ c) {
    const int tid = threadIdx.x;
    const size_t base = (size_t)blockIdx.x * NN;
    const float rv = r[blockIdx.x];
#pragma unroll
    for (int q = 0; q < 8; ++q) {
        const int col = q * 1024 + tid * 4;
        float4 p = *(float4*)(P + base + col);
        const float4 cv = *(const float4*)(c + col);
        p.x *= rv * cv.x;
        p.y *= rv * cv.y;
        p.z *= rv * cv.z;
        p.w *= rv * cv.w;
        *(float4*)(P + base + col) = p;
    }
}

extern "C" void kernel_launch(void* const* d_in, const int* in_sizes, int n_in,
                              void* d_out, int out_size, void* d_ws, size_t ws_size,
                              hipStream_t stream) {
    const float* mat = (const float*)d_in[0];
    float* P = (float*)d_out;

    float* r = (float*)d_ws;   // 8192
    float* c = r + NN;         // 8192
    float* y = c + NN;         // 8192 (row matvec result)
    float* z = y + NN;         // 8192 (col matvec accumulator)

    rowexp_kernel<<<NN, 256, 0, stream>>>(mat, P);
    init_rc<<<NN / 256, 256, 0, stream>>>(r, c);

    for (int it = 0; it < ITERS; ++it) {
        rowmatvec_wmma<<<NN / 16, 256, 0, stream>>>(P, c, y);
        update_r_zero_z<<<NN / 256, 256, 0, stream>>>(r, y, z);
        colmatvec<<<dim3(8, 32), 256, 0, stream>>>(P, r, z);
        update_c<<<NN / 256, 256, 0, stream>>>(c, z);
    }

    final_scale<<<NN, 256, 0, stream>>>(P, r, c);
}